// PoseOptPSO_3977139716274
// MI455X (gfx1250) — compile-verified
//
#include <hip/hip_runtime.h>
#include <math.h>

// ---------------------------------------------------------------------------
// PoseOptPSO forward kinematics + x16 broadcast for MI455X (gfx1250).
//
// Bandwidth analysis: 277 MB of output stores vs 0.05 GFLOP of compute ->
// purely store-bound. Kernel 1 computes the 8192 unique poses (thread per
// chain, LDS-resident l2w transforms). Kernel 2 does the x16 fan-out using
// the CDNA5 async global<->LDS path (ASYNCcnt) when available.
// ---------------------------------------------------------------------------

#define JNT 24

__constant__ int c_parents[JNT] = {-1,0,0,0,1,2,3,4,5,6,7,8,9,9,9,12,13,14,16,17,18,19,20,21};

#if defined(__has_builtin)
#if __has_builtin(__builtin_amdgcn_global_load_async_to_lds_b128) && \
    __has_builtin(__builtin_amdgcn_global_store_async_from_lds_b128) && \
    __has_builtin(__builtin_amdgcn_s_wait_asynccnt)
#define USE_ASYNC_COPY 1
#endif
#endif
#ifndef USE_ASYNC_COPY
#define USE_ASYNC_COPY 0
#endif

#if USE_ASYNC_COPY
// From the hipcc diagnostic: builtin expects pointer-to-int4-vector operands.
typedef int v4i_t __attribute__((vector_size(4 * sizeof(int))));
typedef __attribute__((address_space(1))) v4i_t* g_v4i_ptr;   // global
typedef __attribute__((address_space(3))) v4i_t* l_v4i_ptr;   // LDS
#endif

// ---------------------------------------------------------------------------
// Kernel 1: one thread per unique pose. Computes rvecs (+residual), Rodrigues,
// the serial kinematic chain, pelvis shift, kp and inverse transforms (skts),
// and writes them into the s=0 slot of each skip-row group of d_out.
// LDS: 32 threads x 289 floats (24 joints x 12 floats, padded; stride 289
// => lane stride 33 banks mod 64, coprime -> conflict-free).
// ---------------------------------------------------------------------------
__global__ __launch_bounds__(32) void fk_unique_kernel(
    const float* __restrict__ bones, const float* __restrict__ pelvis,
    const float* __restrict__ particles, const float* __restrict__ rest_pose,
    const int* __restrict__ kp_idxs, const int* __restrict__ particle_idx_p,
    float* __restrict__ out, int skip, int n_particles, int nq, long long Btot)
{
    __shared__ float lds[32 * 289];
    const int u = blockIdx.x * 32 + threadIdx.x;
    if (u >= nq) return;
    float* L = &lds[threadIdx.x * 289];

    const long long b0 = (long long)u * skip;
    const int f = kp_idxs[b0];
    const int pidx = particle_idx_p[0];
    const float* res = particles + (((long long)f * n_particles + pidx) * (JNT + 1)) * 3;
    const float* bn  = bones + b0 * (JNT * 3);

    const long long sk_off = Btot * (long long)(JNT * 3);
    const long long rv_off = Btot * (long long)(JNT * 3 + JNT * 16);

    float* outrv = out + rv_off + b0 * (JNT * 3);
    const float r0x = rest_pose[0], r0y = rest_pose[1], r0z = rest_pose[2];

    for (int j = 0; j < JNT; ++j) {
        const float rx = bn[j*3+0] + res[j*3+0];
        const float ry = bn[j*3+1] + res[j*3+1];
        const float rz = bn[j*3+2] + res[j*3+2];
        outrv[j*3+0] = rx; outrv[j*3+1] = ry; outrv[j*3+2] = rz;

        // Rodrigues, matching reference: R = I + s*K + (1-c)*K^2, K from axis=rv/theta
        const float dot = rx*rx + ry*ry + rz*rz + 1e-12f;
        const float th  = sqrtf(dot);
        const float inv = 1.0f / th;
        const float ax = rx*inv, ay = ry*inv, az = rz*inv;
        const float s = sinf(th), c = cosf(th);
        const float A = 1.0f - c;
        const float n2 = ax*ax + ay*ay + az*az;   // ~1 (eps in theta)
        const float xy = ax*ay, xz = ax*az, yz = ay*az;
        const float R00 = 1.0f + A*(ax*ax - n2);
        const float R01 = -s*az + A*xy;
        const float R02 =  s*ay + A*xz;
        const float R10 =  s*az + A*xy;
        const float R11 = 1.0f + A*(ay*ay - n2);
        const float R12 = -s*ax + A*yz;
        const float R20 = -s*ay + A*xz;
        const float R21 =  s*ax + A*yz;
        const float R22 = 1.0f + A*(az*az - n2);

        float* Lj = L + j * 12;
        if (j == 0) {
            Lj[0]=R00; Lj[1]=R01; Lj[2]=R02; Lj[3]=r0x;
            Lj[4]=R10; Lj[5]=R11; Lj[6]=R12; Lj[7]=r0y;
            Lj[8]=R20; Lj[9]=R21; Lj[10]=R22; Lj[11]=r0z;
        } else {
            const int p = c_parents[j];
            const float tx = rest_pose[j*3+0] - rest_pose[p*3+0];
            const float ty = rest_pose[j*3+1] - rest_pose[p*3+1];
            const float tz = rest_pose[j*3+2] - rest_pose[p*3+2];
            const float* Lp = L + p * 12;
            const float P00=Lp[0],P01=Lp[1],P02=Lp[2], Pt0=Lp[3];
            const float P10=Lp[4],P11=Lp[5],P12=Lp[6], Pt1=Lp[7];
            const float P20=Lp[8],P21=Lp[9],P22=Lp[10],Pt2=Lp[11];
            Lj[0] = P00*R00 + P01*R10 + P02*R20;
            Lj[1] = P00*R01 + P01*R11 + P02*R21;
            Lj[2] = P00*R02 + P01*R12 + P02*R22;
            Lj[3] = P00*tx  + P01*ty  + P02*tz + Pt0;
            Lj[4] = P10*R00 + P11*R10 + P12*R20;
            Lj[5] = P10*R01 + P11*R11 + P12*R21;
            Lj[6] = P10*R02 + P11*R12 + P12*R22;
            Lj[7] = P10*tx  + P11*ty  + P12*tz + Pt1;
            Lj[8] = P20*R00 + P21*R10 + P22*R20;
            Lj[9] = P20*R01 + P21*R11 + P22*R21;
            Lj[10]= P20*R02 + P21*R12 + P22*R22;
            Lj[11]= P20*tx  + P21*ty  + P22*tz + Pt2;
        }
    }

    // pelvis shift: root_locs - rest_pose[0], root_locs = pelvis[f] + residual[24]
    const float shx = pelvis[(long long)f*3+0] + res[JNT*3+0] - r0x;
    const float shy = pelvis[(long long)f*3+1] + res[JNT*3+1] - r0y;
    const float shz = pelvis[(long long)f*3+2] + res[JNT*3+2] - r0z;

    float*  outkp = out + b0 * (JNT * 3);
    float4* outsk = (float4*)(out + sk_off + b0 * (JNT * 16));
    for (int j = 0; j < JNT; ++j) {
        const float* Lj = L + j * 12;
        const float kx = Lj[3] + shx, ky = Lj[7] + shy, kz = Lj[11] + shz;
        outkp[j*3+0]=kx; outkp[j*3+1]=ky; outkp[j*3+2]=kz;
        // Rt = R^T; tinv = -Rt * kp  (Rt[i][b] = R[b][i] = Lj[4*b+i])
        const float t0 = -(Lj[0]*kx + Lj[4]*ky + Lj[8]*kz);
        const float t1 = -(Lj[1]*kx + Lj[5]*ky + Lj[9]*kz);
        const float t2 = -(Lj[2]*kx + Lj[6]*ky + Lj[10]*kz);
        outsk[j*4+0] = make_float4(Lj[0], Lj[4], Lj[8],  t0);
        outsk[j*4+1] = make_float4(Lj[1], Lj[5], Lj[9],  t1);
        outsk[j*4+2] = make_float4(Lj[2], Lj[6], Lj[10], t2);
        outsk[j*4+3] = make_float4(0.f, 0.f, 0.f, 1.f);
    }
}

// ---------------------------------------------------------------------------
// Kernel 2: x(skip) broadcast. One thread owns one unique float4 (read once,
// written to the other skip-1 row slots). Reads hit L2 (17 MB unique data),
// stores are contiguous B128. CDNA5 path: global -> LDS -> global via the
// async data mover (ASYNCcnt), no VGPR round-trip.
// ---------------------------------------------------------------------------
__global__ __launch_bounds__(256) void bcast_kernel(
    float* __restrict__ out, int nq, int skip, long long Btot)
{
#if USE_ASYNC_COPY
    __shared__ float4 stage[256];
#endif
    const int KPu = nq * 18;   // kp:    18 float4 per unique row
    const int SKu = nq * 96;   // skts:  96 float4 per unique row
    const int total = KPu + SKu + KPu;
    const int t = blockIdx.x * 256 + threadIdx.x;
    if (t >= total) return;

    long long base4; int stride4, u, r;
    if (t < KPu)            { base4 = 0;          stride4 = 18; u = t / 18;            r = t % 18; }
    else if (t < KPu + SKu) { base4 = Btot * 18;  stride4 = 96; u = (t - KPu) / 96;    r = (t - KPu) % 96; }
    else                    { base4 = Btot * 114; stride4 = 18; u = (t - KPu - SKu)/18; r = (t - KPu - SKu) % 18; }

    float4* p4 = (float4*)out;
    const long long src = base4 + ((long long)u * skip) * stride4 + r;

#if USE_ASYNC_COPY
    l_v4i_ptr lslot = (l_v4i_ptr)&stage[threadIdx.x];
    __builtin_amdgcn_global_load_async_to_lds_b128(
        (g_v4i_ptr)(p4 + src), lslot, 0, 0);
    __builtin_amdgcn_s_wait_asynccnt(0);   // LDS slot now holds the data
    for (int s = 1; s < skip; ++s) {
        __builtin_amdgcn_global_store_async_from_lds_b128(
            (g_v4i_ptr)(p4 + src + (long long)s * stride4), lslot, 0, 0);
    }
    // s_endpgm performs an implicit wait-idle -> stores drain before exit
#else
    const float4 v = p4[src];
    for (int s = 1; s < skip; ++s) p4[src + (long long)s * stride4] = v;
#endif
}

// ---------------------------------------------------------------------------
// Host side. Inputs (setup_inputs order):
//  0: kp3d (B,24,3) f32 [shape only]   1: bones (B,24,3) f32
//  2: pelvis (F,3) f32                 3: particles (F,P,25,3) f32
//  4: rest_pose (24,3) f32             5: kp_idxs (B,) i32
//  6: N_unique (scalar i32, device)    7: particle_idx (scalar i32, device)
// Output: concat(kp_full B*24*3, skts_full B*24*16, rvecs_full B*24*3) f32.
// ---------------------------------------------------------------------------
extern "C" void kernel_launch(void* const* d_in, const int* in_sizes, int n_in,
                              void* d_out, int out_size, void* d_ws, size_t ws_size,
                              hipStream_t stream) {
    (void)n_in; (void)out_size; (void)d_ws; (void)ws_size;
    const float* bones      = (const float*)d_in[1];
    const float* pelvis     = (const float*)d_in[2];
    const float* particles  = (const float*)d_in[3];
    const float* rest_pose  = (const float*)d_in[4];
    const int*   kp_idxs    = (const int*)d_in[5];
    const int*   pidx_ptr   = (const int*)d_in[7];
    float* out = (float*)d_out;

    const int B = in_sizes[0] / (JNT * 3);          // 131072
    int nq = 8192;                                   // N_UNIQUE (module constant)
    if (B < nq) nq = B;
    const int skip = B / nq;                         // 16
    const int n_frames = in_sizes[2] / 3;            // 10000
    const int n_particles = in_sizes[3] / (n_frames * (JNT + 1) * 3); // 10

    fk_unique_kernel<<<(nq + 31) / 32, 32, 0, stream>>>(
        bones, pelvis, particles, rest_pose, kp_idxs, pidx_ptr,
        out, skip, n_particles, nq, (long long)B);

    const long long total = (long long)nq * 132;     // 18 + 96 + 18 float4 per unique row
    bcast_kernel<<<(int)((total + 255) / 256), 256, 0, stream>>>(
        out, nq, skip, (long long)B);
}